// simCLRLoss_31739808317577
// MI455X (gfx1250) — compile-verified
//
#include <hip/hip_runtime.h>

// CDNA5 / gfx1250, wave32.
typedef __attribute__((ext_vector_type(2))) float v2f;
typedef __attribute__((ext_vector_type(8))) float v8f;

#if defined(__gfx1250__) && __has_builtin(__builtin_amdgcn_wmma_f32_16x16x4_f32)
#define USE_WMMA 1
#else
#define USE_WMMA 0
#endif

#if __has_builtin(__builtin_amdgcn_sched_barrier)
#define SCHED_FENCE() __builtin_amdgcn_sched_barrier(0)
#else
#define SCHED_FENCE()
#endif

__device__ __forceinline__ float softplus_(float x) {
    // numerically stable log(1 + exp(x))
    return fmaxf(x, 0.0f) + log1pf(__expf(-fabsf(x)));
}

#if USE_WMMA
#define WMMA_ACC(acc, prod) \
    acc = __builtin_amdgcn_wmma_f32_16x16x4_f32(false, (prod), false, onesv, \
                                                (short)0, acc, false, false)
#endif

// One workgroup (256 threads = 8 waves) per 16-row tile; waves split K.
// Six per-row sums (3 squared norms + 3 dots) are computed as WMMA row-sums:
// A = elementwise-product tile in the documented 16x4 f32 A-layout,
// B = all-ones (layout independent) -> C[m][*] = rowsum(m) replicated.
__global__ __launch_bounds__(256) void simclr_rowstats_kernel(
    const float* __restrict__ adv, const float* __restrict__ a1,
    const float* __restrict__ a2, float* __restrict__ partial)
{
    constexpr int D = 1024;
    constexpr int kPerWave = D / 8;              // 128 K-values per wave

    const int lane  = threadIdx.x & 31;
    const int wave  = threadIdx.x >> 5;          // 0..7
    const int row16 = lane & 15;                 // row within tile (A-layout)
    const int kcol  = 2 * (lane >> 4);           // lanes 16-31 carry K+2,K+3
    const int rowOff = (blockIdx.x * 16 + row16) * D;
    const int base   = rowOff + wave * kPerWave + kcol;

    const float* p1 = a1  + base;
    const float* p2 = a2  + base;
    const float* pv = adv + base;

    __shared__ float s[8][16][6];                // [wave][row][stat]
    __shared__ float rloss[16];

#if USE_WMMA
    v8f cn1 = {}, cn2 = {}, cnv = {}, c12 = {}, c1v = {}, c2v = {};
    v2f onesv; onesv.x = 1.0f; onesv.y = 1.0f;

    #pragma unroll 1
    for (int it = 0; it < kPerWave / 16; ++it) { // 8 iterations, 4 chunks each
        v2f x1[4], x2[4], xv[4];
        #pragma unroll
        for (int u = 0; u < 4; ++u) {            // 12 loads, imm offsets 0/16/32/48B
            x1[u] = *(const v2f*)(p1 + 4 * u);
            x2[u] = *(const v2f*)(p2 + 4 * u);
            xv[u] = *(const v2f*)(pv + 4 * u);
        }
        // Keep all 12 independent loads in flight before any consumer issues:
        // forces back-to-back global_load_b64 + partial s_wait_loadcnt drains.
        SCHED_FENCE();
        #pragma unroll
        for (int u = 0; u < 4; ++u) {            // 24 WMMAs, 6 interleaved chains
            WMMA_ACC(cn1, x1[u] * x1[u]);
            WMMA_ACC(cn2, x2[u] * x2[u]);
            WMMA_ACC(cnv, xv[u] * xv[u]);
            WMMA_ACC(c12, x1[u] * x2[u]);
            WMMA_ACC(c1v, x1[u] * xv[u]);
            WMMA_ACC(c2v, x2[u] * xv[u]);
        }
        p1 += 16; p2 += 16; pv += 16;
    }

    // C/D layout: VGPR v, lanes 0-15 -> M=v; lanes 16-31 -> M=8+v. Columns are
    // identical (B = ones), so lane 0 / lane 16 carry all 16 row sums.
    if (lane == 0 || lane == 16) {
        const int rbase = (lane == 0) ? 0 : 8;
        #pragma unroll
        for (int v = 0; v < 8; ++v) {
            s[wave][rbase + v][0] = cn1[v];
            s[wave][rbase + v][1] = cn2[v];
            s[wave][rbase + v][2] = cnv[v];
            s[wave][rbase + v][3] = c12[v];
            s[wave][rbase + v][4] = c1v[v];
            s[wave][rbase + v][5] = c2v[v];
        }
    }
#else
    float n1 = 0.f, n2 = 0.f, nv = 0.f, d12 = 0.f, d1v = 0.f, d2v = 0.f;
    #pragma unroll 4
    for (int k = 0; k < kPerWave; k += 4) {
        v2f x1 = *(const v2f*)(p1 + k);
        v2f x2 = *(const v2f*)(p2 + k);
        v2f xv = *(const v2f*)(pv + k);
        n1  = fmaf(x1.x, x1.x, fmaf(x1.y, x1.y, n1));
        n2  = fmaf(x2.x, x2.x, fmaf(x2.y, x2.y, n2));
        nv  = fmaf(xv.x, xv.x, fmaf(xv.y, xv.y, nv));
        d12 = fmaf(x1.x, x2.x, fmaf(x1.y, x2.y, d12));
        d1v = fmaf(x1.x, xv.x, fmaf(x1.y, xv.y, d1v));
        d2v = fmaf(x2.x, xv.x, fmaf(x2.y, xv.y, d2v));
    }
    n1  += __shfl_xor(n1, 16, 32);
    n2  += __shfl_xor(n2, 16, 32);
    nv  += __shfl_xor(nv, 16, 32);
    d12 += __shfl_xor(d12, 16, 32);
    d1v += __shfl_xor(d1v, 16, 32);
    d2v += __shfl_xor(d2v, 16, 32);
    if (lane < 16) {
        s[wave][lane][0] = n1;  s[wave][lane][1] = n2;  s[wave][lane][2] = nv;
        s[wave][lane][3] = d12; s[wave][lane][4] = d1v; s[wave][lane][5] = d2v;
    }
#endif

    __syncthreads();

    // Fixed-order cross-wave sum: 96 threads each own one (row, stat).
    const int t = threadIdx.x;
    if (t < 96) {
        const int r = t / 6, j = t % 6;
        float acc = 0.f;
        #pragma unroll
        for (int w = 0; w < 8; ++w) acc += s[w][r][j];
        s[0][r][j] = acc;
    }
    __syncthreads();

    // Per-row weighted BCE contribution.
    if (t < 16) {
        const float n1_ = s[0][t][0], n2_ = s[0][t][1], nv_ = s[0][t][2];
        const float d12 = s[0][t][3], d1v = s[0][t][4], d2v = s[0][t][5];
        const float EPS  = 1e-12f;
        const float in1  = 1.0f / fmaxf(sqrtf(n1_), EPS);
        const float in2  = 1.0f / fmaxf(sqrtf(n2_), EPS);
        const float inv_ = 1.0f / fmaxf(sqrtf(nv_), EPS);
        const float invT = 1.0f / 0.07f;
        const float l0 = d12 * in1 * in2  * invT;   // label 1, weight 1
        const float l1 = d1v * in1 * inv_ * invT;   // label 0, weight 3
        const float l2 = d2v * in2 * inv_ * invT;   // label 0, weight 3
        rloss[t] = softplus_(-l0) + 3.0f * (softplus_(l1) + softplus_(l2));
    }
    __syncthreads();

    if (t == 0) {
        float acc = 0.f;
        #pragma unroll
        for (int r = 0; r < 16; ++r) acc += rloss[r];
        partial[blockIdx.x] = acc;
    }
}

// Deterministic single-block finalize: fixed-order strided sum + LDS tree.
__global__ __launch_bounds__(256) void simclr_finalize_kernel(
    const float* __restrict__ partial, float* __restrict__ out,
    int nPartial, float scale)
{
    __shared__ float s[256];
    float acc = 0.f;
    for (int i = threadIdx.x; i < nPartial; i += 256) acc += partial[i];
    s[threadIdx.x] = acc;
    __syncthreads();
    #pragma unroll
    for (int stride = 128; stride > 0; stride >>= 1) {
        if (threadIdx.x < (unsigned)stride) s[threadIdx.x] += s[threadIdx.x + stride];
        __syncthreads();
    }
    if (threadIdx.x == 0) out[0] = s[0] * scale;
}

extern "C" void kernel_launch(void* const* d_in, const int* in_sizes, int n_in,
                              void* d_out, int out_size, void* d_ws, size_t ws_size,
                              hipStream_t stream) {
    const float* adv = (const float*)d_in[0];  // adv_pc_logits
    const float* a1  = (const float*)d_in[1];  // aug_pc1_logits
    const float* a2  = (const float*)d_in[2];  // aug_pc2_logits
    const int D = 1024;
    const int N = in_sizes[0] / D;             // 16384
    float* partial = (float*)d_ws;             // N/16 floats of scratch

    const int nBlocks = N / 16;                // one 16-row tile per block
    simclr_rowstats_kernel<<<nBlocks, 256, 0, stream>>>(adv, a1, a2, partial);
    simclr_finalize_kernel<<<1, 256, 0, stream>>>(partial, (float*)d_out,
                                                  nBlocks, 1.0f / (3.0f * (float)N));
}